// RayTransform_17446157156959
// MI455X (gfx1250) — compile-verified
//
#include <hip/hip_runtime.h>
#include <math.h>

typedef float v2f __attribute__((ext_vector_type(2)));
typedef float v8f __attribute__((ext_vector_type(8)));

// ---------------------------------------------------------------------------
// Kernel 1: one thread computes R = exp(K) (Rodrigues) and T = -(trans @ R),
// writes 12 floats to workspace: ws[3k+m] = R[k][m] (k=row, m=col), ws[9+m]=T[m].
// ---------------------------------------------------------------------------
__global__ void rodrigues_precompute(const float* __restrict__ rot_vec,
                                     const float* __restrict__ trans,
                                     float* __restrict__ ws) {
    if (threadIdx.x != 0 || blockIdx.x != 0) return;
    float x = rot_vec[0], y = rot_vec[1], z = rot_vec[2];
    float th2 = x * x + y * y + z * z;
    float th  = sqrtf(th2);
    bool small = th < 1e-8f;
    float st = small ? 1.0f : th;
    float a = small ? (1.0f - th2 / 6.0f)  : (sinf(st) / st);
    float b = small ? (0.5f - th2 / 24.0f) : ((1.0f - cosf(st)) / (st * st));

    float R[3][3];
    R[0][0] = 1.0f - b * (y * y + z * z);
    R[0][1] = -a * z + b * x * y;
    R[0][2] =  a * y + b * x * z;
    R[1][0] =  a * z + b * x * y;
    R[1][1] = 1.0f - b * (x * x + z * z);
    R[1][2] = -a * x + b * y * z;
    R[2][0] = -a * y + b * x * z;
    R[2][1] =  a * x + b * y * z;
    R[2][2] = 1.0f - b * (x * x + y * y);

    float t0 = trans[0], t1 = trans[1], t2 = trans[2];
    for (int k = 0; k < 3; ++k)
        for (int m = 0; m < 3; ++m)
            ws[3 * k + m] = R[k][m];
    for (int m = 0; m < 3; ++m)
        ws[9 + m] = -(t0 * R[0][m] + t1 * R[1][m] + t2 * R[2][m]);
}

// ---------------------------------------------------------------------------
// Kernel 2: one wave handles one 16-ray tile with two V_WMMA_F32_16X16X4_F32
// (one for pos with homogeneous 1.0, one for dir with 0.0). A = [R^T | T] is
// wave-uniform; B columns are the rays; D rows 0..2 are the outputs, already
// lane-major -> fully coalesced b32 loads/stores, no LDS needed.
// ---------------------------------------------------------------------------
__global__ __launch_bounds__(256) void ray_transform_wmma(
    const float* __restrict__ pos, const float* __restrict__ dir,
    const float* __restrict__ ws,
    float* __restrict__ out_pos, float* __restrict__ out_dir, int n_rays) {

    const int lane = threadIdx.x & 31;
    const int wave = (int)((blockIdx.x * blockDim.x + threadIdx.x) >> 5);
    const int ray0 = wave * 16;
    if (ray0 >= n_rays) return;  // wave-uniform: EXEC stays all-1s for WMMA

    // Uniform load of R (row-major) and T from workspace.
    float s0 = ws[0], s1 = ws[1],  s2 = ws[2];
    float s3 = ws[3], s4 = ws[4],  s5 = ws[5];
    float s6 = ws[6], s7 = ws[7],  s8 = ws[8];
    float s9 = ws[9], s10 = ws[10], s11 = ws[11];

    const int  m  = lane & 15;
    const bool hi = lane >= 16;

    // A-matrix (16x4): A[m][k] = R[k][m] for k<3, A[m][3] = T[m]; rows m>=3 are 0.
    // Lane layout: lanes 0-15 hold {K=0,K=1}, lanes 16-31 hold {K=2,K=3}.
    float r0m = (m == 0) ? s0 : (m == 1) ? s1 : (m == 2) ? s2 : 0.0f;  // R[0][m]
    float r1m = (m == 0) ? s3 : (m == 1) ? s4 : (m == 2) ? s5 : 0.0f;  // R[1][m]
    float r2m = (m == 0) ? s6 : (m == 1) ? s7 : (m == 2) ? s8 : 0.0f;  // R[2][m]
    float tm  = (m == 0) ? s9 : (m == 1) ? s10 : (m == 2) ? s11 : 0.0f; // T[m]
    v2f A;
    A.x = hi ? r2m : r0m;
    A.y = hi ? tm  : r1m;

    // B-matrix (4x16): column n = ray n's [x,y,z,h].
    // Lane n (lo) holds {x_n, y_n}; lane n+16 (hi) holds {z_n, h}.
    int r = ray0 + m;
    if (r >= n_rays) r = n_rays - 1;  // clamp (garbage columns masked at store)
    const size_t base = (size_t)r * 3;

    float p0 = pos[base + (hi ? 2 : 0)];  // lo: x, hi: z
    float p1 = pos[base + 1];             // y (unused by hi lanes)
    v2f Bp;
    Bp.x = p0;
    Bp.y = hi ? 1.0f : p1;   // homogeneous 1 -> adds T

    float q0 = dir[base + (hi ? 2 : 0)];
    float q1 = dir[base + 1];
    v2f Bd;
    Bd.x = q0;
    Bd.y = hi ? 0.0f : q1;   // homogeneous 0 -> no translation

    v8f Cz = {};
    // D = A x B + 0  (8 args: neg_a, A, neg_b, B, c_mod, C, reuse_a, reuse_b)
    v8f Dp = __builtin_amdgcn_wmma_f32_16x16x4_f32(
        false, A, false, Bp, (short)0, Cz, false, false);
    v8f Dd = __builtin_amdgcn_wmma_f32_16x16x4_f32(
        false, A, false, Bd, (short)0, Cz, false, false);

    // D layout: VGPR r, lanes 0-15 -> M=r, N=lane. We need M=0..2 for ray N=lane.
    if (lane < 16) {
        int rr = ray0 + lane;
        if (rr < n_rays) {
            float* op = out_pos + (size_t)rr * 3;
            op[0] = Dp[0]; op[1] = Dp[1]; op[2] = Dp[2];
            float* od = out_dir + (size_t)rr * 3;
            od[0] = Dd[0]; od[1] = Dd[1]; od[2] = Dd[2];
        }
    }
}

// ---------------------------------------------------------------------------
// Harness entry point.
//   d_in[0]=pos [N*3], d_in[1]=dir [N*3], d_in[2]=rot_vec [3], d_in[3]=trans [3]
//   d_out = local_pos [N*3] ++ local_dir [N*3]
// ---------------------------------------------------------------------------
extern "C" void kernel_launch(void* const* d_in, const int* in_sizes, int n_in,
                              void* d_out, int out_size, void* d_ws, size_t ws_size,
                              hipStream_t stream) {
    const float* pos     = (const float*)d_in[0];
    const float* dir     = (const float*)d_in[1];
    const float* rot_vec = (const float*)d_in[2];
    const float* trans   = (const float*)d_in[3];

    const int n_rays = in_sizes[0] / 3;
    float* ws      = (float*)d_ws;
    float* out_pos = (float*)d_out;
    float* out_dir = (float*)d_out + (size_t)n_rays * 3;

    rodrigues_precompute<<<1, 32, 0, stream>>>(rot_vec, trans, ws);

    const int n_tiles = (n_rays + 15) / 16;       // one 16-ray tile per wave
    const int threads = 256;                       // 8 waves per block
    const int waves_per_block = threads / 32;
    const int blocks = (n_tiles + waves_per_block - 1) / waves_per_block;
    ray_transform_wmma<<<blocks, threads, 0, stream>>>(
        pos, dir, ws, out_pos, out_dir, n_rays);
}